// ChebNetConv_31404800868550
// MI455X (gfx1250) — compile-verified
//
#include <hip/hip_runtime.h>
#include <hip/hip_bf16.h>

typedef __attribute__((ext_vector_type(2))) float v2f;
typedef __attribute__((ext_vector_type(8))) float v8f;

#define F_DIM 128
#define KCHEB 4

// ---------------------------------------------------------------------------
// Pack W [128, 512] (col index = fin*4 + k) into Bpack[k][fin][fout]
// so GEMM B-fragment loads are coalesced along fout. 256 KB, lives in L2.
// ---------------------------------------------------------------------------
__global__ __launch_bounds__(256) void pack_w_kernel(const float* __restrict__ W,
                                                     float* __restrict__ Bp) {
    int o = blockIdx.x * 256 + threadIdx.x;      // 4*128*128 = 65536 threads
    int fout = o & 127;
    int fin  = (o >> 7) & 127;
    int k    = o >> 14;
    Bp[o] = W[fout * (F_DIM * KCHEB) + fin * KCHEB + k];
}

// T1 = 0 ; T2 = -x      (recurrence term folded into scatter destination)
__global__ __launch_bounds__(256) void init_t1t2_kernel(const float4* __restrict__ x4,
                                                        float4* __restrict__ t1,
                                                        float4* __restrict__ t2, int n4) {
    int i = blockIdx.x * 256 + threadIdx.x;
    if (i >= n4) return;
    float4 v = x4[i];
    t1[i] = make_float4(0.f, 0.f, 0.f, 0.f);
    t2[i] = make_float4(-v.x, -v.y, -v.z, -v.w);
}

// T3 = -T1   (must run after SpMM #1)
__global__ __launch_bounds__(256) void init_t3_kernel(const float4* __restrict__ t1,
                                                      float4* __restrict__ t3, int n4) {
    int i = blockIdx.x * 256 + threadIdx.x;
    if (i >= n4) return;
    float4 v = t1[i];
    t3[i] = make_float4(-v.x, -v.y, -v.z, -v.w);
}

// ---------------------------------------------------------------------------
// SpMM scatter: out[row] += scale * val * x[col], one wave per edge.
// Edge scalars are wave-uniform (readfirstlane -> s_load); each lane gathers a
// coalesced float4 of x[col] and issues 4 hardware f32 atomics (L2-resident).
// ---------------------------------------------------------------------------
__global__ __launch_bounds__(256) void spmm_scatter_kernel(const int* __restrict__ rows,
                                                           const int* __restrict__ cols,
                                                           const float* __restrict__ vals,
                                                           const float* __restrict__ x,
                                                           float* __restrict__ out,
                                                           float scale, int nedges) {
    int e = __builtin_amdgcn_readfirstlane((blockIdx.x * 256 + threadIdx.x) >> 5);
    if (e >= nedges) return;
    int lane = threadIdx.x & 31;
    int r = rows[e];
    int c = cols[e];
    float v = vals[e] * scale;
    float4 xv = ((const float4*)(x + (size_t)c * F_DIM))[lane];
    float* o = out + (size_t)r * F_DIM + lane * 4;
    unsafeAtomicAdd(o + 0, v * xv.x);
    unsafeAtomicAdd(o + 1, v * xv.y);
    unsafeAtomicAdd(o + 2, v * xv.z);
    unsafeAtomicAdd(o + 3, v * xv.w);
}

// ---------------------------------------------------------------------------
// Dense GEMM: out = bias + sum_k T_k @ Wk^T using V_WMMA_F32_16X16X4_F32.
// Workgroup = 16 rows x 128 cols; 8 waves, one 16x16 fp32 tile each.
// A tile staged in LDS with stride 132 (conflict-free ds_load_b64 fragments).
// ---------------------------------------------------------------------------
#define A_STRIDE 132

__device__ __forceinline__ v8f cheb_accum(v8f c, const float* __restrict__ S,
                                          const float* __restrict__ Bk, float* As,
                                          int m0, int n, int tid, int lane) {
    __syncthreads();   // protect LDS reuse across sources
    {
        int r  = tid >> 4;           // 0..15
        int c8 = (tid & 15) << 3;    // 0,8,...,120
        int gr = m0 + r;
        if (gr >= n) gr = n - 1;
        const float4* src = (const float4*)(S + (size_t)gr * F_DIM + c8);
        float4* dst = (float4*)(As + r * A_STRIDE + c8);
        dst[0] = src[0];
        dst[1] = src[1];
    }
    __syncthreads();
    int row = lane & 15;
    int kh  = (lane >> 4) << 1;      // 0 for lanes 0-15, 2 for lanes 16-31
    const float* ap = As + row * A_STRIDE + kh;
#pragma unroll
    for (int fin = 0; fin < F_DIM; fin += 4) {
        v2f a = *(const v2f*)(ap + fin);            // ds_load_b64, conflict-free
        v2f b;
        b.x = Bk[(fin + kh) * F_DIM];               // coalesced (fout-contig)
        b.y = Bk[(fin + kh + 1) * F_DIM];
        c = __builtin_amdgcn_wmma_f32_16x16x4_f32(false, a, false, b,
                                                  (short)0, c, false, false);
    }
    return c;
}

__global__ __launch_bounds__(256) void cheb_gemm_kernel(const float* __restrict__ T0,
                                                        const float* __restrict__ T1,
                                                        const float* __restrict__ T2,
                                                        const float* __restrict__ T3,
                                                        const float* __restrict__ Bpack,
                                                        const float* __restrict__ bias,
                                                        float* __restrict__ out, int n) {
    __shared__ float As[16 * A_STRIDE];
    int tid   = threadIdx.x;
    int lane  = tid & 31;
    int wave  = tid >> 5;            // 0..7 -> 16-col slice
    int m0    = blockIdx.x << 4;
    int fout0 = wave << 4;
    int nlo   = lane & 15;

    v8f c = {};
    const float* B0 = Bpack + fout0 + nlo;
    c = cheb_accum(c, T0, B0 + 0 * (F_DIM * F_DIM), As, m0, n, tid, lane);
    c = cheb_accum(c, T1, B0 + 1 * (F_DIM * F_DIM), As, m0, n, tid, lane);
    c = cheb_accum(c, T2, B0 + 2 * (F_DIM * F_DIM), As, m0, n, tid, lane);
    c = cheb_accum(c, T3, B0 + 3 * (F_DIM * F_DIM), As, m0, n, tid, lane);

    int col = fout0 + nlo;
    float bv = bias[col];
    int rbase = m0 + ((lane >> 4) << 3);   // C layout: lanes16-31 hold M=v+8
#pragma unroll
    for (int v = 0; v < 8; ++v) {
        int rr = rbase + v;
        if (rr < n) out[(size_t)rr * F_DIM + col] = c[v] + bv;
    }
}

// ---------------------------------------------------------------------------
extern "C" void kernel_launch(void* const* d_in, const int* in_sizes, int n_in,
                              void* d_out, int out_size, void* d_ws, size_t ws_size,
                              hipStream_t stream) {
    const float* x    = (const float*)d_in[0];
    const int*   rows = (const int*)d_in[1];
    const int*   cols = (const int*)d_in[2];
    const float* vals = (const float*)d_in[3];
    const float* W    = (const float*)d_in[4];
    const float* b    = (const float*)d_in[5];
    float* out = (float*)d_out;

    int n  = in_sizes[0] / F_DIM;     // 100000
    int ne = in_sizes[1];             // 3200000

    size_t tsz = (size_t)n * F_DIM * sizeof(float);
    char* ws = (char*)d_ws;
    float* T1 = (float*)(ws);
    float* T2 = (float*)(ws + tsz);
    float* T3 = (float*)(ws + 2 * tsz);
    float* Bp = (float*)(ws + 3 * tsz);

    // 1) pack W for coalesced B fragments
    pack_w_kernel<<<(KCHEB * F_DIM * F_DIM) / 256, 256, 0, stream>>>(W, Bp);

    int n4 = (int)(((size_t)n * F_DIM) / 4);
    int ib = (n4 + 255) / 256;
    int sb = (int)(((size_t)ne * 32 + 255) / 256);

    // 2) T1 = 0, T2 = -T0
    init_t1t2_kernel<<<ib, 256, 0, stream>>>((const float4*)x, (float4*)T1, (float4*)T2, n4);
    // 3) T1 += L @ T0
    spmm_scatter_kernel<<<sb, 256, 0, stream>>>(rows, cols, vals, x, T1, 1.0f, ne);
    // 4) T3 = -T1
    init_t3_kernel<<<ib, 256, 0, stream>>>((const float4*)T1, (float4*)T3, n4);
    // 5) T2 += 2 L @ T1  -> T2 = 2 L T1 - T0
    spmm_scatter_kernel<<<sb, 256, 0, stream>>>(rows, cols, vals, T1, T2, 2.0f, ne);
    // 6) T3 += 2 L @ T2  -> T3 = 2 L T2 - T1
    spmm_scatter_kernel<<<sb, 256, 0, stream>>>(rows, cols, vals, T2, T3, 2.0f, ne);
    // 7) out = bias + sum_k T_k @ Wk^T   (fp32 WMMA)
    cheb_gemm_kernel<<<(n + 15) / 16, 256, 0, stream>>>(x, T1, T2, T3, Bp, b, out, n);
}